// optimizer_18322330485153
// MI455X (gfx1250) — compile-verified
//
#include <hip/hip_runtime.h>

typedef __attribute__((ext_vector_type(2))) float v2f;
typedef __attribute__((ext_vector_type(8))) float v8f;

#define MU 1e-3f
#define RO 1e-3f

// Problem geometry (from reference): W=8, B=128, S=25, G=240
constexpr int S  = 25;    // inner small dim
constexpr int G  = 240;   // rows of Omega
constexpr int SP = 28;    // S padded to multiple of 4 (K padding for WMMA)
constexpr int NP = 32;    // S padded to multiple of 16 (N padding for WMMA)

__global__ __launch_bounds__(256) void admm_wmma_kernel(
    const float* __restrict__ Yp,       // (S,G) = (25,240)
    const float* __restrict__ Uk,       // (WB,S,S)
    const float* __restrict__ Lk,       // (WB,S)
    const float* __restrict__ Omega0,   // (WB,G,S)
    const int*   __restrict__ nIter,
    float*       __restrict__ out)      // (WB,G,S)
{
    // Y_p^T stored as [g][s], s padded 25->28 with zeros (serves both GEMMs)
    __shared__ float sYpT[G * SP];      // 26880 B
    // Omega as [g][t], t padded 25->32 with zeros (B operand of constraint GEMM)
    __shared__ float sOmega[G * NP];    // 30720 B
    // lambda as [s][t], rows 25..27 and cols 25..31 stay zero
    __shared__ float sLam[SP * NP];     //  3584 B
    __shared__ float sUL[S * NP];       //  3200 B
    __shared__ float sInv[G];           //   960 B   (total ~65.3 KB)

    const int tid  = threadIdx.x;
    const int wb   = blockIdx.x;
    const int wave = tid >> 5;
    const int lane = tid & 31;
    const int half = lane >> 4;   // 0: lanes 0-15, 1: lanes 16-31
    const int l16  = lane & 15;

    // ---------------- init LDS ----------------
    for (int i = tid; i < G * SP; i += 256) {
        int g = i / SP, s = i - g * SP;
        sYpT[i] = (s < S) ? Yp[s * G + g] : 0.0f;
    }
    const float* om0 = Omega0 + (size_t)wb * G * S;
    for (int i = tid; i < G * NP; i += 256) {
        int g = i / NP, t = i - g * NP;
        sOmega[i] = (t < S) ? om0[g * S + t] : 0.0f;
    }
    for (int i = tid; i < SP * NP; i += 256) sLam[i] = 0.0f;
    const float* uk = Uk + (size_t)wb * S * S;
    const float* lk = Lk + (size_t)wb * S;
    for (int i = tid; i < S * NP; i += 256) {
        int s = i / NP, t = i - s * NP;
        sUL[i] = (t < S) ? uk[s * S + t] * lk[t] : 0.0f;
    }
    __syncthreads();

    const int iters = nIter[0];

    if (wave < 4) {
        // ================= waves 0..3: constraint GEMM path =================
        // C2(25x25) = Y_p(25x240) @ Omega(240x25), one 16x16 tile per wave.
        // A = Y_p is loop-invariant: hoist all 60 K-step fragments to VGPRs.
        const int m0 = (wave >> 1) * 16;
        const int n0 = (wave & 1) * 16;
        const int mi = m0 + l16;
        const bool mv = (mi < SP);          // rows 25..27 read the zero pad

        v2f aC[60];
        #pragma unroll
        for (int ks = 0; ks < 60; ++ks) {
            const int kk = 4 * ks + 2 * half;
            aC[ks].x = mv ? sYpT[kk * SP + mi] : 0.0f;        // A[m][kk]
            aC[ks].y = mv ? sYpT[(kk + 1) * SP + mi] : 0.0f;  // A[m][kk+1]
        }

        for (int it = 0; it < iters; ++it) {
            if (tid < G) {                  // rows 0..127 of the norm
                float s2 = 0.0f;
                #pragma unroll
                for (int t = 0; t < S; ++t) {
                    float v = sOmega[tid * NP + t];
                    s2 += v * v;
                }
                sInv[tid] = 1.0f / sqrtf(s2);
            }

            v8f c = (v8f){};
            #pragma unroll
            for (int ks = 0; ks < 60; ++ks) {
                const int kk = 4 * ks + 2 * half;
                v2f b;
                b.x = sOmega[kk * NP + n0 + l16];
                b.y = sOmega[(kk + 1) * NP + n0 + l16];
                c = __builtin_amdgcn_wmma_f32_16x16x4_f32(
                    false, aC[ks], false, b, (short)0, c, false, false);
            }

            __syncthreads();                // barrier 1

            // lam += RO * (C2 - UL) on valid (M<25, N<25) elements
            const int N = n0 + l16;
            if (N < S) {
                #pragma unroll
                for (int r = 0; r < 8; ++r) {
                    const int M = m0 + r + 8 * half;
                    if (M < S) {
                        const int idx = M * NP + N;
                        sLam[idx] += RO * (c[r] - sUL[idx]);
                    }
                }
            }
            __syncthreads();                // barrier 2
        }
    } else {
        // ================= waves 4..7: grad GEMM path =================
        // C1(240x25) = Y_p^T(240x25) @ lam(25x25); wave owns up to 4 mtiles,
        // each with both n-tiles. A = Y_p^T hoisted; B shared across mtiles.
        const int w4 = wave - 4;

        v2f aG[4][7] = {};
        #pragma unroll
        for (int i = 0; i < 4; ++i) {
            const int mt = w4 + 4 * i;      // waves cover mtiles 0..14
            if (mt < 15) {
                const int mrow = mt * 16 + l16;
                #pragma unroll
                for (int ks = 0; ks < 7; ++ks) {
                    const int kk = 4 * ks + 2 * half;
                    aG[i][ks].x = sYpT[mrow * SP + kk];
                    aG[i][ks].y = sYpT[mrow * SP + kk + 1];
                }
            }
        }

        for (int it = 0; it < iters; ++it) {
            if (tid < G) {                  // rows 128..239 of the norm
                float s2 = 0.0f;
                #pragma unroll
                for (int t = 0; t < S; ++t) {
                    float v = sOmega[tid * NP + t];
                    s2 += v * v;
                }
                sInv[tid] = 1.0f / sqrtf(s2);
            }

            // B = lam fragments: shared across all 4 mtiles of this wave
            v2f bG[2][7];
            #pragma unroll
            for (int nt = 0; nt < 2; ++nt) {
                #pragma unroll
                for (int ks = 0; ks < 7; ++ks) {
                    const int kk = 4 * ks + 2 * half;
                    bG[nt][ks].x = sLam[kk * NP + nt * 16 + l16];
                    bG[nt][ks].y = sLam[(kk + 1) * NP + nt * 16 + l16];
                }
            }

            v8f acc[4][2];
            #pragma unroll
            for (int i = 0; i < 4; ++i)
                #pragma unroll
                for (int nt = 0; nt < 2; ++nt)
                    acc[i][nt] = (v8f){};

            #pragma unroll
            for (int i = 0; i < 4; ++i) {
                const int mt = w4 + 4 * i;
                if (mt < 15) {
                    #pragma unroll
                    for (int nt = 0; nt < 2; ++nt) {
                        #pragma unroll
                        for (int ks = 0; ks < 7; ++ks) {
                            acc[i][nt] = __builtin_amdgcn_wmma_f32_16x16x4_f32(
                                false, aG[i][ks], false, bG[nt][ks],
                                (short)0, acc[i][nt], false, false);
                        }
                    }
                }
            }

            __syncthreads();                // barrier 1 (sInv also ready)

            // Omega -= MU * (Omega * inv_norm + C1)
            #pragma unroll
            for (int i = 0; i < 4; ++i) {
                const int mt = w4 + 4 * i;
                if (mt < 15) {
                    #pragma unroll
                    for (int nt = 0; nt < 2; ++nt) {
                        const int N = nt * 16 + l16;
                        if (N < S) {
                            #pragma unroll
                            for (int r = 0; r < 8; ++r) {
                                const int M = mt * 16 + r + 8 * half;
                                const int idx = M * NP + N;
                                const float o = sOmega[idx];
                                sOmega[idx] = o - MU * (o * sInv[M] + acc[i][nt][r]);
                            }
                        }
                    }
                }
            }
            __syncthreads();                // barrier 2
        }
    }

    // ---------------- writeout ----------------
    float* dst = out + (size_t)wb * G * S;
    for (int i = tid; i < G * S; i += 256) {
        int g = i / S, t = i - g * S;
        dst[i] = sOmega[g * NP + t];
    }
}

extern "C" void kernel_launch(void* const* d_in, const int* in_sizes, int n_in,
                              void* d_out, int out_size, void* d_ws, size_t ws_size,
                              hipStream_t stream) {
    const float* Yp     = (const float*)d_in[0];  // (25,240)
    const float* Uk     = (const float*)d_in[1];  // (W,B,25,25)
    const float* Lk     = (const float*)d_in[2];  // (W,B,25)
    const float* Omega0 = (const float*)d_in[3];  // (W,B,240,25)
    const int*   nIter  = (const int*)d_in[4];    // scalar
    float* out = (float*)d_out;

    const int blocks = in_sizes[3] / (G * S);     // W*B = 1024
    admm_wmma_kernel<<<blocks, 256, 0, stream>>>(Yp, Uk, Lk, Omega0, nIter, out);
}